// GATBlock_5918464934677
// MI455X (gfx1250) — compile-verified
//
#include <hip/hip_runtime.h>
#include <math.h>

// ---------------------------------------------------------------------------
// GAT block for MI455X (gfx1250, wave32).
//   N nodes, IN=OUT=128, H=4 heads, C=32 channels/head, E edges + N self loops
// ---------------------------------------------------------------------------

typedef float v2f __attribute__((ext_vector_type(2)));
typedef float v8f __attribute__((ext_vector_type(8)));

#define INF_ 128
#define HEADS 4
#define NEG_SLOPE 0.2f

// Monotone encoding of float into uint so unsigned atomicMax == float max.
__device__ __forceinline__ unsigned f32_enc(float f) {
    unsigned u = __float_as_uint(f);
    return (u & 0x80000000u) ? ~u : (u | 0x80000000u);
}
__device__ __forceinline__ float f32_dec(unsigned u) {
    return (u & 0x80000000u) ? __uint_as_float(u & 0x7FFFFFFFu)
                             : __uint_as_float(~u);
}

__device__ __forceinline__ float leaky(float e) {
    return e > 0.0f ? e : NEG_SLOPE * e;
}

// Native no-return f32 atomic add (GLOBAL_ATOMIC_ADD_F32, STOREcnt-tracked).
__device__ __forceinline__ void atom_add_f32(float* p, float v) {
    unsafeAtomicAdd(p, v);
}

// ---------------------------------------------------------------------------
// K1: x_proj = x @ Wg^T ; skip = x @ Ws^T  via V_WMMA_F32_16X16X4_F32.
// blockDim = (32, 8): wave y computes output columns [16y, 16y+16) for one
// 16-row strip of x. K=128 -> 32 WMMA steps of K=4.
//
// A (16x4, f32, 2 VGPRs): lanes 0-15 hold M=lane, K={0,1}; lanes 16-31 hold
//   M=lane-16, K={2,3}   (ISA 7.12.2).
// B (4x16, f32, 2 VGPRs): lanes 0-15 hold N=lane with rows K={0,1} in v0/v1;
//   lanes 16-31 hold rows K={2,3}. B[k][n] = W[n][k] (row-major W, contiguous
//   over k -> float2 loads).
// C/D (16x16 f32, 8 VGPRs): VGPR j holds row j (lanes 0-15) / j+8 (lanes16-31).
// ---------------------------------------------------------------------------
__global__ void __launch_bounds__(256)
gemm_proj_kernel(const float* __restrict__ x,
                 const float* __restrict__ Wg,
                 const float* __restrict__ Ws,
                 float* __restrict__ xproj,
                 float* __restrict__ skip,
                 int n) {
    const int rowBase = blockIdx.x * 16;
    const int colBase = threadIdx.y * 16;          // 0,16,...,112
    const int lane    = threadIdx.x;               // 0..31
    const int m       = lane & 15;
    const int koff    = (lane >> 4) * 2;           // 0 or 2

    int arow = rowBase + m;
    if (arow >= n) arow = n - 1;                   // clamp loads, keep EXEC full
    const float* __restrict__ xr = x  + (size_t)arow * INF_;
    const float* __restrict__ gr = Wg + (size_t)(colBase + m) * INF_;
    const float* __restrict__ sr = Ws + (size_t)(colBase + m) * INF_;

    v8f accG = {};
    v8f accS = {};
#pragma unroll
    for (int k0 = 0; k0 < INF_; k0 += 4) {
        v2f a  = *(const v2f*)(xr + k0 + koff);
        v2f bg = *(const v2f*)(gr + k0 + koff);
        v2f bs = *(const v2f*)(sr + k0 + koff);
        accG = __builtin_amdgcn_wmma_f32_16x16x4_f32(
            false, a, false, bg, (short)0, accG, false, false);
        accS = __builtin_amdgcn_wmma_f32_16x16x4_f32(
            false, a, false, bs, (short)0, accS, false, false);
    }

    const int rOff = (lane >> 4) * 8;
#pragma unroll
    for (int j = 0; j < 8; ++j) {
        const int r = rowBase + rOff + j;
        if (r < n) {
            xproj[(size_t)r * INF_ + colBase + m] = accG[j];
            skip [(size_t)r * INF_ + colBase + m] = accS[j];
        }
    }
}

// ---------------------------------------------------------------------------
// K2: per-(node,head) attention scalars: a_src/a_dst = <x_proj[n,h,:], att>
// ---------------------------------------------------------------------------
__global__ void att_kernel(const float* __restrict__ xproj,
                           const float* __restrict__ att_src,
                           const float* __restrict__ att_dst,
                           float* __restrict__ a_src,
                           float* __restrict__ a_dst,
                           int n) {
    const int t = blockIdx.x * blockDim.x + threadIdx.x;
    if (t >= n * HEADS) return;
    const int node = t >> 2;
    const int h    = t & 3;
    const float* __restrict__ xp = xproj + (size_t)node * INF_ + h * 32;
    const float* __restrict__ as = att_src + h * 32;
    const float* __restrict__ ad = att_dst + h * 32;
    float s = 0.0f, d = 0.0f;
#pragma unroll
    for (int c = 0; c < 32; ++c) {
        const float v = xp[c];
        s = fmaf(v, as[c], s);
        d = fmaf(v, ad[c], d);
    }
    a_src[t] = s;
    a_dst[t] = d;
}

// ---------------------------------------------------------------------------
// K3: zero-init agg (d_out) and softmax stats. Deterministic each call.
// ---------------------------------------------------------------------------
__global__ void init_kernel(float* __restrict__ out,
                            unsigned* __restrict__ segmax,
                            float* __restrict__ segsum,
                            int n) {
    const int t = blockIdx.x * blockDim.x + threadIdx.x;
    if (t < n * INF_) out[t] = 0.0f;
    if (t < n * HEADS) { segmax[t] = 0u; segsum[t] = 0.0f; }
}

__device__ __forceinline__ void edge_endpoints(const int* __restrict__ ei,
                                               int t, int e,
                                               int& s, int& d) {
    if (t < e) { s = ei[t]; d = ei[e + t]; }
    else       { s = t - e; d = t - e; }          // self loop
}

// K4: segment max of leaky-relu'd scores (encoded-uint atomicMax).
__global__ void edge_max_kernel(const int* __restrict__ ei,
                                const float* __restrict__ a_src,
                                const float* __restrict__ a_dst,
                                unsigned* __restrict__ segmax,
                                int e, int n) {
    const int t = blockIdx.x * blockDim.x + threadIdx.x;
    if (t >= e + n) return;
    int s, d; edge_endpoints(ei, t, e, s, d);
#pragma unroll
    for (int h = 0; h < HEADS; ++h) {
        const float ev = leaky(a_src[s * HEADS + h] + a_dst[d * HEADS + h]);
        atomicMax(&segmax[d * HEADS + h], f32_enc(ev));
    }
}

// K5: segment sum of exp(e - max).
__global__ void edge_sum_kernel(const int* __restrict__ ei,
                                const float* __restrict__ a_src,
                                const float* __restrict__ a_dst,
                                const unsigned* __restrict__ segmax,
                                float* __restrict__ segsum,
                                int e, int n) {
    const int t = blockIdx.x * blockDim.x + threadIdx.x;
    if (t >= e + n) return;
    int s, d; edge_endpoints(ei, t, e, s, d);
#pragma unroll
    for (int h = 0; h < HEADS; ++h) {
        const float ev = leaky(a_src[s * HEADS + h] + a_dst[d * HEADS + h]);
        const float mx = f32_dec(segmax[d * HEADS + h]);
        atom_add_f32(&segsum[d * HEADS + h], __expf(ev - mx));
    }
}

// ---------------------------------------------------------------------------
// K6: wave-per-edge scatter. Lane owns channels [4*lane, 4*lane+4) of the
// 128-wide row (head = lane>>3); gathers x_proj[src] as float4, scales by its
// head's alpha, atomic-adds into agg (= d_out). agg fits in L2 (25.6 MB of
// 192 MB) so the f32 atomics resolve on-chip; no-return adds ride STOREcnt.
// ---------------------------------------------------------------------------
__global__ void __launch_bounds__(256)
edge_scatter_kernel(const int* __restrict__ ei,
                    const float* __restrict__ a_src,
                    const float* __restrict__ a_dst,
                    const unsigned* __restrict__ segmax,
                    const float* __restrict__ segsum,
                    const float* __restrict__ xproj,
                    float* __restrict__ out,
                    int e, int n) {
    const long long gid = (long long)blockIdx.x * blockDim.x + threadIdx.x;
    const int t    = (int)(gid >> 5);
    const int lane = (int)(gid & 31);
    if (t >= e + n) return;
    int s, d; edge_endpoints(ei, t, e, s, d);

    const int h  = lane >> 3;
    const float ev = leaky(a_src[s * HEADS + h] + a_dst[d * HEADS + h]);
    const float mx = f32_dec(segmax[d * HEADS + h]);
    const float sm = segsum[d * HEADS + h];
    const float alpha = __expf(ev - mx) / (sm + 1e-16f);

    const float4 v = *(const float4*)(xproj + (size_t)s * INF_ + lane * 4);
    float* __restrict__ o = out + (size_t)d * INF_ + lane * 4;
    atom_add_f32(o + 0, alpha * v.x);
    atom_add_f32(o + 1, alpha * v.y);
    atom_add_f32(o + 2, alpha * v.z);
    atom_add_f32(o + 3, alpha * v.w);
}

// K7: out = elu(agg + bias + skip)
__global__ void final_kernel(float* __restrict__ out,
                             const float* __restrict__ skip,
                             const float* __restrict__ bias,
                             int n) {
    const int t = blockIdx.x * blockDim.x + threadIdx.x;
    if (t >= n * INF_) return;
    const float v = out[t] + bias[t & (INF_ - 1)] + skip[t];
    out[t] = v > 0.0f ? v : (__expf(v) - 1.0f);
}

// ---------------------------------------------------------------------------
extern "C" void kernel_launch(void* const* d_in, const int* in_sizes, int n_in,
                              void* d_out, int out_size, void* d_ws, size_t ws_size,
                              hipStream_t stream) {
    const float* x    = (const float*)d_in[0];
    const int*   ei   = (const int*)  d_in[1];
    const float* Wg   = (const float*)d_in[2];
    const float* asw  = (const float*)d_in[3];
    const float* adw  = (const float*)d_in[4];
    const float* bias = (const float*)d_in[5];
    const float* Ws   = (const float*)d_in[6];
    float* out = (float*)d_out;

    const int n = in_sizes[0] / INF_;   // 50000
    const int e = in_sizes[1] / 2;      // 800000
    const int etot = e + n;

    // workspace layout (floats)
    float* ws      = (float*)d_ws;
    float* xproj   = ws;                                   // n*128
    float* skip    = xproj + (size_t)n * INF_;             // n*128
    float* a_src   = skip  + (size_t)n * INF_;             // n*4
    float* a_dst   = a_src + (size_t)n * HEADS;            // n*4
    unsigned* smax = (unsigned*)(a_dst + (size_t)n * HEADS); // n*4
    float* ssum    = (float*)(smax + (size_t)n * HEADS);   // n*4

    const dim3 gemmBlock(32, 8);
    gemm_proj_kernel<<<(n + 15) / 16, gemmBlock, 0, stream>>>(
        x, Wg, Ws, xproj, skip, n);

    att_kernel<<<(n * HEADS + 255) / 256, 256, 0, stream>>>(
        xproj, asw, adw, a_src, a_dst, n);

    init_kernel<<<(n * INF_ + 255) / 256, 256, 0, stream>>>(
        out, smax, ssum, n);

    edge_max_kernel<<<(etot + 255) / 256, 256, 0, stream>>>(
        ei, a_src, a_dst, smax, e, n);

    edge_sum_kernel<<<(etot + 255) / 256, 256, 0, stream>>>(
        ei, a_src, a_dst, smax, ssum, e, n);

    const long long scatterThreads = (long long)etot * 32;
    edge_scatter_kernel<<<(unsigned)((scatterThreads + 255) / 256), 256, 0, stream>>>(
        ei, a_src, a_dst, smax, ssum, xproj, out, e, n);

    final_kernel<<<(n * INF_ + 255) / 256, 256, 0, stream>>>(
        out, skip, bias, n);
}